// NodeProcessor_2070174236990
// MI455X (gfx1250) — compile-verified
//
#include <hip/hip_runtime.h>
#include <hip/hip_bf16.h>

typedef float v8f __attribute__((ext_vector_type(8)));
typedef float v2f __attribute__((ext_vector_type(2)));

#define LN_EPS          1e-5f
#define A_STRIDE        260   // 256 + 4 pad; 260 % 64 == 4 -> conflict-free frag loads
#define M_TILE          32    // nodes per wave = two 16-row WMMA tiles
#define WAVES_PER_BLOCK 8
#define SMEM_BYTES      (WAVES_PER_BLOCK * M_TILE * A_STRIDE * 4)  // 266,240 B < 320 KB

#if __has_builtin(__builtin_amdgcn_wmma_f32_16x16x4_f32)
#define HAVE_WMMA_F32 1
#else
#define HAVE_WMMA_F32 0
#endif

// ---------------------------------------------------------------------------
// Stage 2: scatter-sum edge_attr into agg (== d_out) via f32 L2 atomics.
// 32 lanes per edge, each lane handles 4 contiguous floats (float4 load).
// ---------------------------------------------------------------------------
__global__ __launch_bounds__(256) void scatter_edges_kernel(
    const float* __restrict__ edge_attr,
    const long long* __restrict__ dst_idx,   // edge_index row 1 (int64)
    float* agg, int nEdges)
{
  const long long t = (long long)blockIdx.x * blockDim.x + threadIdx.x;
  const int edge  = (int)(t >> 5);
  const int chunk = ((int)t & 31) * 4;
  if (edge >= nEdges) return;
  const int d = (int)dst_idx[edge];
  const float4 v = *(const float4*)(edge_attr + (size_t)edge * 128 + chunk);
  float* p = agg + (size_t)d * 128 + chunk;
  unsafeAtomicAdd(p + 0, v.x);
  unsafeAtomicAdd(p + 1, v.y);
  unsafeAtomicAdd(p + 2, v.z);
  unsafeAtomicAdd(p + 3, v.w);
}

// ---------------------------------------------------------------------------
// One MLP layer: D[32x128] = relu?(act[32xK] @ W[Kx128] + b)
// A frags from wave-private LDS (ds_load_b64), B frags from global (L2-hot).
// f32 WMMA layout: A lane: M=lane%16, K=v+2*(lane/16); B lane: N=lane%16,
// same K mapping; C/D: VGPR r <-> row r+8*(lane/16), lane%16 <-> col.
// ---------------------------------------------------------------------------
template <int K, bool RELU>
__device__ __forceinline__ void run_layer(const float* __restrict__ W,
                                          const float* __restrict__ bias,
                                          const float* act, int nl, int hi,
                                          v8f acc[2][8])
{
  for (int nt = 0; nt < 8; ++nt) {
    v8f a0 = {0.f, 0.f, 0.f, 0.f, 0.f, 0.f, 0.f, 0.f};
    v8f a1 = {0.f, 0.f, 0.f, 0.f, 0.f, 0.f, 0.f, 0.f};
    const int col = nt * 16 + nl;
#if HAVE_WMMA_F32
#pragma unroll 4
    for (int k0 = 0; k0 < K; k0 += 4) {
      const int kb = k0 + 2 * hi;
      v2f bf;
      bf.x = W[(size_t)kb * 128 + col];
      bf.y = W[(size_t)(kb + 1) * 128 + col];
      const v2f af0 = *(const v2f*)(act + (0 * 16 + nl) * A_STRIDE + kb);
      const v2f af1 = *(const v2f*)(act + (1 * 16 + nl) * A_STRIDE + kb);
      a0 = __builtin_amdgcn_wmma_f32_16x16x4_f32(false, af0, false, bf,
                                                 (short)0, a0, false, false);
      a1 = __builtin_amdgcn_wmma_f32_16x16x4_f32(false, af1, false, bf,
                                                 (short)0, a1, false, false);
    }
#else
    // Scalar fallback (should not be hit on gfx1250): each lane computes its
    // own D elements directly from LDS + global.
    for (int r = 0; r < 8; ++r) {
      float s0 = 0.f, s1 = 0.f;
      const int row0 = r + 8 * hi;
      for (int k = 0; k < K; ++k) {
        const float w = W[(size_t)k * 128 + col];
        s0 += act[(0 * 16 + row0) * A_STRIDE + k] * w;
        s1 += act[(1 * 16 + row0) * A_STRIDE + k] * w;
      }
      a0[r] = s0; a1[r] = s1;
    }
#endif
    const float bv = bias[col];
#pragma unroll
    for (int r = 0; r < 8; ++r) {
      a0[r] += bv; a1[r] += bv;
      if (RELU) { a0[r] = fmaxf(a0[r], 0.f); a1[r] = fmaxf(a1[r], 0.f); }
    }
    acc[0][nt] = a0;
    acc[1][nt] = a1;
  }
}

// Write accumulators back into the wave-private LDS activation buffer
// (stride 260 keeps the next layer's ds_load_b64 frags bank-conflict-free).
__device__ __forceinline__ void stage_acts(float* act, int nl, int hi,
                                           const v8f acc[2][8])
{
  asm volatile("s_wait_dscnt 0" ::: "memory");
#pragma unroll
  for (int mt = 0; mt < 2; ++mt)
#pragma unroll
    for (int nt = 0; nt < 8; ++nt)
#pragma unroll
      for (int r = 0; r < 8; ++r)
        act[(mt * 16 + r + 8 * hi) * A_STRIDE + nt * 16 + nl] = acc[mt][nt][r];
  asm volatile("s_wait_dscnt 0" ::: "memory");
}

// ---------------------------------------------------------------------------
// Stage 3: fused concat -> MLP(256->128->128->128) -> LayerNorm.
// One wave per 32-node tile; 8 waves / 256-thread block; 266 KB LDS / block.
// agg aliases out (each wave reads its agg rows into LDS before the final
// store overwrites them; rows are exclusively owned).
// ---------------------------------------------------------------------------
__global__ __launch_bounds__(256) void node_mlp_ln_kernel(
    const float* __restrict__ x, const float* agg,
    const float* __restrict__ W1, const float* __restrict__ b1,
    const float* __restrict__ W2, const float* __restrict__ b2,
    const float* __restrict__ W3, const float* __restrict__ b3,
    const float* __restrict__ lnw, const float* __restrict__ lnb,
    float* out, int nNodes)
{
  extern __shared__ float smem[];
  const int wave = threadIdx.x >> 5;
  const int lane = threadIdx.x & 31;
  const int nl   = lane & 15;
  const int hi   = lane >> 4;
  float* act = smem + wave * (M_TILE * A_STRIDE);

  const int nTiles = (nNodes + M_TILE - 1) / M_TILE;
  const int tile   = blockIdx.x * WAVES_PER_BLOCK + wave;
  if (tile >= nTiles) return;  // wave-uniform: surviving waves keep EXEC all-1s
  const int base = tile * M_TILE;

  // Stage [x | agg] tile (32 rows x 256 cols) into LDS, coalesced float4 loads.
  for (int idx = lane; idx < M_TILE * 32; idx += 32) {
    const int row = idx >> 5;
    const int c4  = idx & 31;
    int gr = base + row;
    if (gr >= nNodes) gr = nNodes - 1;
    const float4 vx = ((const float4*)(x   + (size_t)gr * 128))[c4];
    const float4 va = ((const float4*)(agg + (size_t)gr * 128))[c4];
    *(float4*)(act + row * A_STRIDE + c4 * 4)       = vx;
    *(float4*)(act + row * A_STRIDE + 128 + c4 * 4) = va;
  }
  asm volatile("s_wait_dscnt 0" ::: "memory");

  v8f acc[2][8];
  run_layer<256, true >(W1, b1, act, nl, hi, acc);
  stage_acts(act, nl, hi, acc);
  run_layer<128, true >(W2, b2, act, nl, hi, acc);
  stage_acts(act, nl, hi, acc);
  run_layer<128, false>(W3, b3, act, nl, hi, acc);

  // LayerNorm across the 128 features of each row. Each 16-lane half holds
  // 8 full rows spread over lanes (cols) and VGPRs (rows): butterfly-reduce.
  for (int mt = 0; mt < 2; ++mt) {
    float s[8], q[8];
#pragma unroll
    for (int r = 0; r < 8; ++r) { s[r] = 0.f; q[r] = 0.f; }
#pragma unroll
    for (int nt = 0; nt < 8; ++nt)
#pragma unroll
      for (int r = 0; r < 8; ++r) {
        const float v = acc[mt][nt][r];
        s[r] += v; q[r] += v * v;
      }
    for (int off = 1; off < 16; off <<= 1)
#pragma unroll
      for (int r = 0; r < 8; ++r) {
        s[r] += __shfl_xor(s[r], off, 32);
        q[r] += __shfl_xor(q[r], off, 32);
      }
    float mu[8], rs[8];
#pragma unroll
    for (int r = 0; r < 8; ++r) {
      mu[r] = s[r] * (1.f / 128.f);
      const float var = q[r] * (1.f / 128.f) - mu[r] * mu[r];
      rs[r] = rsqrtf(var + LN_EPS);
    }
    for (int nt = 0; nt < 8; ++nt) {
      const float lw = lnw[nt * 16 + nl];
      const float lb = lnb[nt * 16 + nl];
#pragma unroll
      for (int r = 0; r < 8; ++r) {
        const int row = base + mt * 16 + r + 8 * hi;
        if (row < nNodes)
          out[(size_t)row * 128 + nt * 16 + nl] =
              (acc[mt][nt][r] - mu[r]) * rs[r] * lw + lb;
      }
    }
  }
}

// ---------------------------------------------------------------------------
// Host launcher. Inputs (setup_inputs order):
// 0:x 1:edge_index(int64) 2:edge_attr 3:W1 4:b1 5:W2 6:b2 7:W3 8:b3 9:ln_w 10:ln_b
// ---------------------------------------------------------------------------
extern "C" void kernel_launch(void* const* d_in, const int* in_sizes, int n_in,
                              void* d_out, int out_size, void* d_ws, size_t ws_size,
                              hipStream_t stream)
{
  const float*     x   = (const float*)d_in[0];
  const long long* ei  = (const long long*)d_in[1];
  const float*     ea  = (const float*)d_in[2];
  const float*     W1  = (const float*)d_in[3];
  const float*     b1  = (const float*)d_in[4];
  const float*     W2  = (const float*)d_in[5];
  const float*     b2  = (const float*)d_in[6];
  const float*     W3  = (const float*)d_in[7];
  const float*     b3  = (const float*)d_in[8];
  const float*     lnw = (const float*)d_in[9];
  const float*     lnb = (const float*)d_in[10];
  float* out = (float*)d_out;

  const int nNodes = in_sizes[0] / 128;
  const int nEdges = in_sizes[1] / 2;
  const long long* dst = ei + nEdges;  // edge_index[1] = destination nodes

  // Stage 1: agg accumulator lives in d_out (exact size); zero it.
  hipMemsetAsync(out, 0, (size_t)nNodes * 128 * sizeof(float), stream);

  // Stage 2: scatter-sum via global_atomic_add_f32.
  {
    const long long threads = (long long)nEdges * 32;
    const int blocks = (int)((threads + 255) / 256);
    scatter_edges_kernel<<<blocks, 256, 0, stream>>>(ea, dst, out, nEdges);
  }

  // Stage 3: fused MLP + LayerNorm (WMMA f32).
  {
    (void)hipFuncSetAttribute((const void*)node_mlp_ln_kernel,
                              hipFuncAttributeMaxDynamicSharedMemorySize,
                              SMEM_BYTES);
    const int nTiles = (nNodes + M_TILE - 1) / M_TILE;
    const int blocks = (nTiles + WAVES_PER_BLOCK - 1) / WAVES_PER_BLOCK;
    node_mlp_ln_kernel<<<blocks, 256, SMEM_BYTES, stream>>>(
        x, out, W1, b1, W2, b2, W3, b3, lnw, lnb, out, nNodes);
  }
}